// RandomCentroidLabelAffineTransform3D_48163763257561
// MI455X (gfx1250) — compile-verified
//
#include <hip/hip_runtime.h>
#include <hip/hip_bf16.h>

typedef __attribute__((ext_vector_type(2))) float v2f;
typedef __attribute__((ext_vector_type(8))) float v8f;

#define DSZ 144
#define D2  (144 * 144)
#define D3  (144 * 144 * 144)
#define NB  4
#define KCH 5   // ceil(144/32) k-chunks of 32 per (i,j) row

// ---------------------------------------------------------------------------
// Kernel 1: closed-form solve of the normal equations for theta (2 pairs x B).
// A = old^T old = [[4,0,0,-2],[0,4,0,-2],[0,0,4,-2],[-2,-2,-2,4]]
//   => x3 = b3 + (b0+b1+b2)/2 ; xi = bi/4 + x3/2
// theta stored as th[(pair*NB+b)*12 + q*3 + p]
// ---------------------------------------------------------------------------
__global__ void theta_setup_kernel(const float* __restrict__ noise_m,
                                   const float* __restrict__ noise_f,
                                   float* __restrict__ theta) {
    int t = threadIdx.x;
    if (t >= 24) return;            // 2 pairs * 4 batches * 3 columns
    int pair = t / 12;
    int r    = t % 12;
    int b    = r / 3;
    int p    = r % 3;

    const float OLD[4][4] = {
        {-1.f, -1.f, -1.f, 1.f},
        {-1.f, -1.f,  1.f, 1.f},
        {-1.f,  1.f, -1.f, 1.f},
        { 1.f, -1.f, -1.f, 1.f}};

    const float* nz = pair ? noise_f : noise_m;
    float np[4];
#pragma unroll
    for (int q = 0; q < 4; ++q)
        np[q] = 1.0f - 0.1f * nz[b * 12 + q * 3 + p];   // 1 - SCALE*noise

    float bv[4];
#pragma unroll
    for (int i = 0; i < 4; ++i) {
        float s = 0.f;
#pragma unroll
        for (int q = 0; q < 4; ++q)
            s += OLD[q][i] * OLD[q][p] * np[q];         // b = old^T @ (old[:,:3]*np)
        bv[i] = s;
    }
    float x3 = bv[3] + 0.5f * (bv[0] + bv[1] + bv[2]);
    float* th = theta + (pair * NB + b) * 12;
    th[0 * 3 + p] = 0.25f * bv[0] + 0.5f * x3;
    th[1 * 3 + p] = 0.25f * bv[1] + 0.5f * x3;
    th[2 * 3 + p] = 0.25f * bv[2] + 0.5f * x3;
    th[3 * 3 + p] = x3;
}

// ---------------------------------------------------------------------------
// Kernel 2: fused warp_grid (via V_WMMA_F32_16X16X4_F32) + trilinear resample
// of image+label pair sharing the same grid. One wave handles 32 consecutive
// k voxels of one (pair,b,i,j) row.
//   A rows 0..2   = theta^T           (coords of (i,j,k0+n),      lanes 0-15)
//   A rows 8..10  = theta^T + 16*tk   (coords of (i,j,k0+16+n),   lanes 16-31)
//   B cols n      = (i, j, k0+n, 1)
// => lane L ends up holding coords of voxel (i, j, k0+L) in D[0..2].
// ---------------------------------------------------------------------------
__global__ __launch_bounds__(256) void resample_pair_kernel(
    const float* __restrict__ mov_img, const float* __restrict__ fix_img,
    const float* __restrict__ mov_lbl, const float* __restrict__ fix_lbl,
    const float* __restrict__ theta,   float* __restrict__ out) {

    const int lane = threadIdx.x & 31;
    const int wave = threadIdx.x >> 5;
    long chunk = (long)blockIdx.x * 8 + wave;            // wave-uniform
    const long totalChunks = (long)2 * NB * DSZ * DSZ * KCH;
    if (chunk >= totalChunks) return;                    // whole-wave exit

    int  kc   = (int)(chunk % KCH); long tmp = chunk / KCH;
    int  j    = (int)(tmp % DSZ);   tmp /= DSZ;
    int  i    = (int)(tmp % DSZ);   tmp /= DSZ;
    int  b    = (int)(tmp % NB);
    int  pair = (int)(tmp / NB);
    int  k0   = kc * 32;

    const float* th = theta + (pair * NB + b) * 12;      // th[q*3+p]

    // ---- Build A (16x4 f32 operand: 2 VGPRs; ISA 7.12.2 layout) ----
    int  m     = lane & 15;
    bool hi    = lane >= 16;                              // K=2/3 half
    bool valid = (m < 3) | ((m >= 8) & (m < 11));
    int  pp    = (m < 3) ? m : ((m >= 8 && m < 11) ? (m - 8) : 0);
    float sel  = valid ? 1.0f : 0.0f;
    float t0 = th[0 * 3 + pp], t1 = th[1 * 3 + pp];
    float t2 = th[2 * 3 + pp], t3 = th[3 * 3 + pp];
    float koff = (m >= 8) ? 16.0f : 0.0f;                 // affine k+16 shift
    v2f A;
    A.x = sel * (hi ? t2 : t0);                           // K0 / K2
    A.y = sel * (hi ? (t3 + koff * t2) : t1);             // K1 / K3

    // ---- Build B (4x16 f32 operand: 2 VGPRs, rows striped across lanes) ----
    v2f Bm;
    Bm.x = hi ? (float)(k0 + lane - 16) : (float)i;       // row K0=i / K2=k
    Bm.y = hi ? 1.0f : (float)j;                          // row K1=j / K3=1

    v8f C = {};
    v8f Dm = __builtin_amdgcn_wmma_f32_16x16x4_f32(
        /*neg_a=*/false, A, /*neg_b=*/false, Bm,
        /*c_mod=*/(short)0, C, /*reuse_a=*/false, /*reuse_b=*/false);

    float gx = Dm[0], gy = Dm[1], gz = Dm[2];             // grid coords

    // ---- Trilinear sample with boundary clamp (matches jnp.clip) ----
    float fx = floorf(gx), fy = floorf(gy), fz = floorf(gz);
    float wx = gx - fx,    wy = gy - fy,    wz = gz - fz;
    const float dmax = (float)(DSZ - 1);
    int ix0 = (int)fminf(fmaxf(fx,        0.f), dmax);
    int iy0 = (int)fminf(fmaxf(fy,        0.f), dmax);
    int iz0 = (int)fminf(fmaxf(fz,        0.f), dmax);
    int ix1 = (int)fminf(fmaxf(fx + 1.f,  0.f), dmax);
    int iy1 = (int)fminf(fmaxf(fy + 1.f,  0.f), dmax);
    int iz1 = (int)fminf(fmaxf(fz + 1.f,  0.f), dmax);

    const float* img = (pair ? fix_img : mov_img) + (size_t)b * D3;
    const float* lbl = (pair ? fix_lbl : mov_lbl) + (size_t)b * D3;

    int x0o = ix0 * D2,  x1o = ix1 * D2;
    int y0o = iy0 * DSZ, y1o = iy1 * DSZ;
    int o000 = x0o + y0o + iz0, o001 = x0o + y0o + iz1;
    int o010 = x0o + y1o + iz0, o011 = x0o + y1o + iz1;
    int o100 = x1o + y0o + iz0, o101 = x1o + y0o + iz1;
    int o110 = x1o + y1o + iz0, o111 = x1o + y1o + iz1;

    // 16 gathers: image + label share indices & weights (grid fusion)
    float a000 = img[o000], a001 = img[o001], a010 = img[o010], a011 = img[o011];
    float a100 = img[o100], a101 = img[o101], a110 = img[o110], a111 = img[o111];
    float l000 = lbl[o000], l001 = lbl[o001], l010 = lbl[o010], l011 = lbl[o011];
    float l100 = lbl[o100], l101 = lbl[o101], l110 = lbl[o110], l111 = lbl[o111];

    // factored lerp: z, then y, then x (7 fma-lerps per volume)
    float a00 = fmaf(wz, a001 - a000, a000), a01 = fmaf(wz, a011 - a010, a010);
    float a10 = fmaf(wz, a101 - a100, a100), a11 = fmaf(wz, a111 - a110, a110);
    float a0  = fmaf(wy, a01  - a00,  a00),  a1  = fmaf(wy, a11  - a10,  a10);
    float av  = fmaf(wx, a1   - a0,   a0);

    float l00 = fmaf(wz, l001 - l000, l000), l01 = fmaf(wz, l011 - l010, l010);
    float l10 = fmaf(wz, l101 - l100, l100), l11 = fmaf(wz, l111 - l110, l110);
    float l0  = fmaf(wy, l01  - l00,  l00),  l1  = fmaf(wy, l11  - l10,  l10);
    float lv  = fmaf(wx, l1   - l0,   l0);

    int k = k0 + lane;
    if (k < DSZ) {                                        // mask tail chunk
        int vox    = i * D2 + j * DSZ + k;
        int ch_img = pair ? 1 : 0;                        // out[0]=mov_img, [1]=fix_img
        int ch_lbl = pair ? 3 : 2;                        // out[2]=mov_lbl, [3]=fix_lbl
        out[(size_t)(ch_img * NB + b) * D3 + vox] = av;   // coalesced along k
        out[(size_t)(ch_lbl * NB + b) * D3 + vox] = lv;
    }
}

extern "C" void kernel_launch(void* const* d_in, const int* in_sizes, int n_in,
                              void* d_out, int out_size, void* d_ws, size_t ws_size,
                              hipStream_t stream) {
    const float* mov_img = (const float*)d_in[0];
    const float* fix_img = (const float*)d_in[1];
    const float* mov_lbl = (const float*)d_in[2];
    const float* fix_lbl = (const float*)d_in[3];
    // d_in[4] = indices (unused by the reference computation)
    const float* noise_m = (const float*)d_in[5];
    const float* noise_f = (const float*)d_in[6];
    float* out   = (float*)d_out;
    float* theta = (float*)d_ws;    // 2 * 4 * 12 floats

    theta_setup_kernel<<<1, 32, 0, stream>>>(noise_m, noise_f, theta);

    const long totalChunks = (long)2 * NB * DSZ * DSZ * KCH;   // 829440 waves
    const int  blocks      = (int)((totalChunks + 7) / 8);     // 8 waves / block
    resample_pair_kernel<<<blocks, 256, 0, stream>>>(
        mov_img, fix_img, mov_lbl, fix_lbl, theta, out);
}